// Network_16587163698006
// MI455X (gfx1250) — compile-verified
//
#include <hip/hip_runtime.h>
#include <hip/hip_bf16.h>
#include <cstdint>

// ROI crop (bilinear, align_corners=True, zeros padding) + 2x2 max pool.
// bottom: [1,512,38,50] f32, rois: [1024,5] f32, out: [1024,512,7,7] f32.

#define NROIS 1024
#define CCH   512
#define FH    38
#define FW    50
#define PS    7
#define PRE   14          // pooling_size * 2
#define CB    8           // channels staged per block
#define CHW   (FH*FW)     // 1900 floats per channel slice
#define THREADS 256

__device__ __forceinline__ float iy_of(float y1, float y2, int k) {
    // reference: t11=(y2-y1)/(H-1); t12=(y1+y2-H+1)/(H-1); base=linspace(-1,1,PRE)
    const float inv = 1.0f / (float)(FH - 1);
    float t11 = (y2 - y1) * inv;
    float t12 = (y1 + y2 - (float)(FH - 1)) * inv;
    float base = -1.0f + (float)k * (2.0f / (float)(PRE - 1));
    float gy = t11 * base + t12;
    return (gy + 1.0f) * 0.5f * (float)(FH - 1);
}

__global__ __launch_bounds__(THREADS)
void roi_crop_pool_kernel(const float* __restrict__ bottom,
                          const float* __restrict__ rois,
                          float* __restrict__ out)
{
    __shared__ __align__(16) float s_feat[CB * CHW];   // 15200 floats (60800 B) worst case
    __shared__ int   s_xi[2][PRE];
    __shared__ float s_xw[2][PRE];
    __shared__ int   s_yo[2][PRE];   // (clamped_row - ylo) * FW
    __shared__ float s_yw[2][PRE];

    const int r   = blockIdx.x;
    const int c0  = blockIdx.y * CB;
    const int tid = threadIdx.x;

    // ---- uniform per-block ROI y-window (rows actually touched) ----
    const float y1 = rois[r * 5 + 2] * 0.0625f;
    const float y2 = rois[r * 5 + 4] * 0.0625f;
    int ylo = (int)floorf(iy_of(y1, y2, 0));
    int yhi = (int)floorf(iy_of(y1, y2, PRE - 1)) + 1;
    ylo = min(max(ylo, 0), FH - 1);
    yhi = min(max(yhi, 0), FH - 1);
    const int NR   = yhi - ylo + 1;      // rows staged per channel
    const int nr25 = NR * 25;            // b64 transfers per channel (NR*50 floats)

    // ---- async DMA: stage CB channels x NR rows into LDS (ASYNCcnt) ----
    const uint32_t lds_base = (uint32_t)(uintptr_t)(&s_feat[0]);  // low 32b of flat = LDS offset
    for (int c = 0; c < CB; ++c) {
        const float* gsrc = bottom + (size_t)(c0 + c) * CHW + (size_t)ylo * FW;
        const uint32_t lch = lds_base + (uint32_t)(c * NR * FW * 4);
        for (int t = tid; t < nr25; t += THREADS) {
            uint32_t laddr = lch + (uint32_t)(t * 8);
            uint64_t gaddr = (uint64_t)(uintptr_t)(gsrc + t * 2);
            asm volatile("global_load_async_to_lds_b64 %0, %1, off"
                         :: "v"(laddr), "v"(gaddr) : "memory");
        }
    }

    // ---- per-ROI sampling tables (overlaps with async DMA in flight) ----
    if (tid < PRE) {
        const int k = tid;
        float x1 = rois[r * 5 + 1] * 0.0625f;
        float x2 = rois[r * 5 + 3] * 0.0625f;
        const float inv = 1.0f / (float)(FW - 1);
        float t00 = (x2 - x1) * inv;
        float t02 = (x1 + x2 - (float)(FW - 1)) * inv;
        float base = -1.0f + (float)k * (2.0f / (float)(PRE - 1));
        float ix  = ((t00 * base + t02) + 1.0f) * 0.5f * (float)(FW - 1);
        float xf0 = floorf(ix);
        float ax  = ix - xf0;
        int xi0 = (int)xf0, xi1 = xi0 + 1;
        bool v0 = (xi0 >= 0) && (xi0 <= FW - 1);
        bool v1 = (xi1 >= 0) && (xi1 <= FW - 1);
        s_xi[0][k] = min(max(xi0, 0), FW - 1);
        s_xi[1][k] = min(max(xi1, 0), FW - 1);
        s_xw[0][k] = v0 ? 1.0f - ax : 0.0f;
        s_xw[1][k] = v1 ? ax : 0.0f;
    } else if (tid >= 32 && tid < 32 + PRE) {
        const int k = tid - 32;
        float iy  = iy_of(y1, y2, k);
        float yf0 = floorf(iy);
        float ay  = iy - yf0;
        int yi0 = (int)yf0, yi1 = yi0 + 1;
        bool v0 = (yi0 >= 0) && (yi0 <= FH - 1);
        bool v1 = (yi1 >= 0) && (yi1 <= FH - 1);
        s_yo[0][k] = (min(max(yi0, 0), FH - 1) - ylo) * FW;
        s_yo[1][k] = (min(max(yi1, 0), FH - 1) - ylo) * FW;
        s_yw[0][k] = v0 ? 1.0f - ay : 0.0f;
        s_yw[1][k] = v1 ? ay : 0.0f;
    }

    asm volatile("s_wait_asynccnt 0" ::: "memory");
    __syncthreads();

    // ---- gather from LDS, bilinear, 2x2 max, streaming NT store ----
    float* dst = out + ((size_t)r * CCH + c0) * (PS * PS);
    for (int o = tid; o < CB * PS * PS; o += THREADS) {
        int c    = o / (PS * PS);
        int cell = o - c * (PS * PS);
        int i    = cell / PS;
        int j    = cell - i * PS;
        const float* f = &s_feat[c * NR * FW];

        int pxA = 2 * j, pxB = pxA + 1, pyA = 2 * i, pyB = pyA + 1;
        int   xA0 = s_xi[0][pxA], xA1 = s_xi[1][pxA];
        float wA0 = s_xw[0][pxA], wA1 = s_xw[1][pxA];
        int   xB0 = s_xi[0][pxB], xB1 = s_xi[1][pxB];
        float wB0 = s_xw[0][pxB], wB1 = s_xw[1][pxB];
        int   yA0 = s_yo[0][pyA], yA1 = s_yo[1][pyA];
        float uA0 = s_yw[0][pyA], uA1 = s_yw[1][pyA];
        int   yB0 = s_yo[0][pyB], yB1 = s_yo[1][pyB];
        float uB0 = s_yw[0][pyB], uB1 = s_yw[1][pyB];

        float vAA = uA0 * (wA0 * f[yA0 + xA0] + wA1 * f[yA0 + xA1])
                  + uA1 * (wA0 * f[yA1 + xA0] + wA1 * f[yA1 + xA1]);
        float vAB = uA0 * (wB0 * f[yA0 + xB0] + wB1 * f[yA0 + xB1])
                  + uA1 * (wB0 * f[yA1 + xB0] + wB1 * f[yA1 + xB1]);
        float vBA = uB0 * (wA0 * f[yB0 + xA0] + wA1 * f[yB0 + xA1])
                  + uB1 * (wA0 * f[yB1 + xA0] + wA1 * f[yB1 + xA1]);
        float vBB = uB0 * (wB0 * f[yB0 + xB0] + wB1 * f[yB0 + xB1])
                  + uB1 * (wB0 * f[yB1 + xB0] + wB1 * f[yB1 + xB1]);

        float m = fmaxf(fmaxf(vAA, vAB), fmaxf(vBA, vBB));
        __builtin_nontemporal_store(m, &dst[o]);
    }
}

extern "C" void kernel_launch(void* const* d_in, const int* in_sizes, int n_in,
                              void* d_out, int out_size, void* d_ws, size_t ws_size,
                              hipStream_t stream) {
    const float* bottom = (const float*)d_in[0];   // [1,512,38,50] f32
    const float* rois   = (const float*)d_in[1];   // [1024,5] f32
    float* out = (float*)d_out;                    // [1024,512,7,7] f32
    (void)in_sizes; (void)n_in; (void)d_ws; (void)ws_size; (void)out_size;

    dim3 grid(NROIS, CCH / CB);
    roi_crop_pool_kernel<<<grid, THREADS, 0, stream>>>(bottom, rois, out);
}